// GNN_ComiRec_SA_SIMRec_68083821576412
// MI455X (gfx1250) — compile-verified
//
#include <hip/hip_runtime.h>
#include <hip/hip_bf16.h>
#include <math.h>

// ---- problem constants (from reference) ----
#define BN      6400      // B*N = 128*50
#define DIM     64
#define KNOISE  1024
#define NORM_TERM 13.815510557964274f   // log(1e6)
#define LOGK      6.931471805599453f    // log(1024)

typedef float v2f __attribute__((ext_vector_type(2)));
typedef float v8f __attribute__((ext_vector_type(8)));

// Fast stable softplus: max(x,0) + log(1 + exp(-|x|)) using HW transcendentals
// (v_exp_f32 / v_log_f32 TRANS ops co-execute with the WMMA/XDL pipe).
__device__ __forceinline__ float softplus_f(float x) {
    return fmaxf(x, 0.0f) + __logf(1.0f + __expf(-fabsf(x)));
}

// ---------------------------------------------------------------------------
// 1) Gather noise embeddings into dense [K,64] + lnq[k] = logprob_noise[noise[k]]
// ---------------------------------------------------------------------------
__global__ void nce_prep(const int* __restrict__ noise,
                         const float* __restrict__ embs,
                         const float* __restrict__ logprob_noise,
                         float* __restrict__ noiseG,
                         float* __restrict__ lnq) {
    int idx = blockIdx.x * blockDim.x + threadIdx.x;
    if (idx < KNOISE * DIM) {
        int k = idx >> 6;
        int d = idx & 63;
        int id = noise[k];
        noiseG[idx] = embs[(size_t)id * DIM + d];
        if (d == 0) lnq[k] = logprob_noise[id];
    }
}

// ---------------------------------------------------------------------------
// 2) Target term: one wave32 per row; softplus(-x) partials per block
// ---------------------------------------------------------------------------
__global__ void nce_target(const int* __restrict__ target,
                           const float* __restrict__ input,
                           const float* __restrict__ embs,
                           const float* __restrict__ logprob_noise,
                           float* __restrict__ partials) {
    __shared__ float wsum[8];
    const int lane = threadIdx.x & 31;
    const int wave = threadIdx.x >> 5;
    const int row  = blockIdx.x * 8 + wave;

    const int tgt = target[row];
    const float* ir = input + (size_t)row * DIM;
    const float* er = embs  + (size_t)tgt * DIM;
    float d = ir[lane] * er[lane] + ir[lane + 32] * er[lane + 32];
    #pragma unroll
    for (int o = 16; o > 0; o >>= 1) d += __shfl_down(d, o, 32);
    if (lane == 0) {
        float x = d - NORM_TERM - logprob_noise[tgt] - LOGK;
        wsum[wave] = softplus_f(-x);
    }
    __syncthreads();
    if (threadIdx.x == 0) {
        float b = 0.0f;
        #pragma unroll
        for (int w = 0; w < 8; ++w) b += wsum[w];
        partials[blockIdx.x] = b;
    }
}

// ---------------------------------------------------------------------------
// 3) Noise GEMM + fused BCE epilogue via V_WMMA_F32_16X16X4_F32.
//    Wave tile: 16 rows x 64 cols (4 accumulator tiles, A-fragments reused).
//    B fragments preloaded into distinct registers -> clause of 16 loads,
//    one wait, 16 back-to-back WMMAs. Next tile's B rows prefetched.
//    Block: 8 waves -> 16 rows x 512 cols. Grid: (400 M-tiles, 2 col groups).
// ---------------------------------------------------------------------------
__global__ void nce_noise_gemm(const float* __restrict__ input,
                               const float* __restrict__ noiseG,
                               const float* __restrict__ lnq,
                               float* __restrict__ partials) {
    __shared__ float wsum[8];
    const int lane = threadIdx.x & 31;
    const int wave = threadIdx.x >> 5;
    const int half = lane >> 4;     // which 16-lane half of the wave
    const int l16  = lane & 15;

    const int m0      = blockIdx.x * 16;
    const int colBase = blockIdx.y * 512 + wave * 64;

    // Preload A fragments for all 16 K-steps (f32 16x4 layout):
    // lane m = l16 ; VGPR pair holds K = j*4 + 2*half + {0,1}
    v2f aF[16];
    const float* arow = input + (size_t)(m0 + l16) * DIM;
    #pragma unroll
    for (int j = 0; j < 16; ++j)
        aF[j] = *(const v2f*)(arow + j * 4 + 2 * half);

    float s = 0.0f;
    #pragma unroll
    for (int t = 0; t < 4; ++t) {
        const int n0 = colBase + t * 16;
        const float* brow = noiseG + (size_t)(n0 + l16) * DIM;  // B[k][n] = noiseG[n][k]

        // Prefetch next tile's B row (global_prefetch_b8) while this tile runs
        if (t < 3)
            __builtin_prefetch(noiseG + (size_t)(n0 + 16 + l16) * DIM, 0, 1);

        // Preload ALL B fragments into distinct registers (clauseable loads)
        v2f bF[16];
        #pragma unroll
        for (int j = 0; j < 16; ++j)
            bF[j] = *(const v2f*)(brow + j * 4 + 2 * half);

        v8f acc = {0.f, 0.f, 0.f, 0.f, 0.f, 0.f, 0.f, 0.f};
        #pragma unroll
        for (int j = 0; j < 16; ++j) {
            // (neg_a, A, neg_b, B, c_mod, C, reuse_a, reuse_b)
            acc = __builtin_amdgcn_wmma_f32_16x16x4_f32(
                false, aF[j], false, bF[j], (short)0, acc, false, false);
        }

        // Epilogue: acc element v is D[m=v+8*half][n=l16]; q depends only on col
        const float q = lnq[n0 + l16];
        #pragma unroll
        for (int v = 0; v < 8; ++v) {
            float x = acc[v] - NORM_TERM - q - LOGK;
            s += softplus_f(x);     // label 0 -> softplus(x)
        }
    }

    // wave reduce, then block reduce -> deterministic partial
    #pragma unroll
    for (int o = 16; o > 0; o >>= 1) s += __shfl_down(s, o, 32);
    if (lane == 0) wsum[wave] = s;
    __syncthreads();
    if (threadIdx.x == 0) {
        float b = 0.0f;
        #pragma unroll
        for (int w = 0; w < 8; ++w) b += wsum[w];
        partials[blockIdx.y * gridDim.x + blockIdx.x] = b;
    }
}

// ---------------------------------------------------------------------------
// 4) Final fixed-order reduction of 1600 partials -> mean loss scalar
// ---------------------------------------------------------------------------
__global__ void nce_reduce(const float* __restrict__ partials, int n,
                           float* __restrict__ out) {
    __shared__ float sm[256];
    float s = 0.0f;
    for (int i = threadIdx.x; i < n; i += 256) s += partials[i];
    sm[threadIdx.x] = s;
    __syncthreads();
    for (int o = 128; o > 0; o >>= 1) {
        if ((int)threadIdx.x < o) sm[threadIdx.x] += sm[threadIdx.x + o];
        __syncthreads();
    }
    if (threadIdx.x == 0) out[0] = sm[0] / (float)BN;
}

// ---------------------------------------------------------------------------
extern "C" void kernel_launch(void* const* d_in, const int* in_sizes, int n_in,
                              void* d_out, int out_size, void* d_ws, size_t ws_size,
                              hipStream_t stream) {
    (void)in_sizes; (void)n_in; (void)out_size; (void)ws_size;
    const int*   target = (const int*)d_in[0];
    const float* input  = (const float*)d_in[1];
    const float* embs   = (const float*)d_in[2];
    const int*   noise  = (const int*)d_in[3];
    const float* lpn    = (const float*)d_in[4];

    float* ws     = (float*)d_ws;
    float* noiseG = ws;                         // 65536 floats (256 KB)
    float* lnq    = noiseG + KNOISE * DIM;      // 1024 floats
    float* partA  = lnq + KNOISE;               // 800 floats (gemm partials)
    float* partB  = partA + 800;                // 800 floats (target partials)
    // partA..partB are contiguous: reduce over 1600

    nce_prep<<<(KNOISE * DIM + 255) / 256, 256, 0, stream>>>(noise, embs, lpn,
                                                             noiseG, lnq);
    nce_target<<<BN / 8, 256, 0, stream>>>(target, input, embs, lpn, partB);
    dim3 g(BN / 16, KNOISE / 512);
    nce_noise_gemm<<<g, 256, 0, stream>>>(input, noiseG, lnq, partA);
    nce_reduce<<<1, 256, 0, stream>>>(partA, 1600, (float*)d_out);
}